// CDKANLayer_64355789963717
// MI455X (gfx1250) — compile-verified
//
#include <hip/hip_runtime.h>
#include <hip/hip_bf16.h>

typedef __attribute__((ext_vector_type(2))) float v2f;
typedef __attribute__((ext_vector_type(8))) float v8f;

constexpr int B_   = 256;   // batch
constexpr int S_   = 512;   // sequence length
constexpr int I_   = 128;   // input features
constexpr int O_   = 128;   // output features
constexpr int LAGS = 11;    // MAX_LAG + 1
constexpr int CN   = 8;     // GRID_SIZE + ORDER
constexpr int H_   = 16;    // MOD_H
constexpr float GRIDM1 = 4.0f;  // GRID_SIZE - 1

constexpr int ICHUNKS = 4;          // split i-reduction for WGP coverage
constexpr int IPC     = I_ / ICHUNKS;  // 32 i per chunk, 4 per wave
constexpr int TILES   = (B_ / 16) * (O_ / 16);  // 128

__device__ __forceinline__ float fast_rcp(float x) {
  return __builtin_amdgcn_rcpf(x);
}

#if __has_builtin(__builtin_amdgcn_tanhf)
// gfx1250 native v_tanh_f32: one TRANS op per tanh
__device__ __forceinline__ float fast_tanh(float x) {
  return __builtin_amdgcn_tanhf(x);
}
__device__ __forceinline__ float fast_sigmoid(float x) {
  // sigmoid(x) = 0.5*tanh(x/2) + 0.5  -> single TRANS op
  return fmaf(0.5f, __builtin_amdgcn_tanhf(0.5f * x), 0.5f);
}
#else
__device__ __forceinline__ float fast_tanh(float x) {
  // tanh(x) = 1 - 2/(e^(2x)+1); saturates correctly to +/-1
  return fmaf(-2.0f, fast_rcp(__expf(2.0f * x) + 1.0f), 1.0f);
}
__device__ __forceinline__ float fast_sigmoid(float x) {
  return fast_rcp(1.0f + __expf(-x));
}
#endif

// ---------------------------------------------------------------------------
// Kernel 1: xm[b,i] = mean_s x_history[b,s,i]   (HBM-bound, coalesced in i)
// ---------------------------------------------------------------------------
__global__ __launch_bounds__(I_) void xm_kernel(const float* __restrict__ x,
                                                float* __restrict__ xm) {
  const int b = blockIdx.x;
  const int i = threadIdx.x;
  const float* p = x + (size_t)b * S_ * I_ + i;
  float acc = 0.0f;
#pragma unroll 8
  for (int s = 0; s < S_; ++s) acc += p[(size_t)s * I_];
  xm[b * I_ + i] = acc * (1.0f / (float)S_);
}

// ---------------------------------------------------------------------------
// Kernel 2: w_lag = softmax(lag_logits) over last axis, one (o,i) per thread
// ---------------------------------------------------------------------------
__global__ __launch_bounds__(128) void softmax_kernel(const float* __restrict__ logits,
                                                      float* __restrict__ wlag) {
  const int p = blockIdx.x * blockDim.x + threadIdx.x;  // 0 .. O*I-1
  const float* lp = logits + (size_t)p * LAGS;
  float v[LAGS];
  float m = lp[0];
#pragma unroll
  for (int l = 0; l < LAGS; ++l) { v[l] = lp[l]; m = fmaxf(m, v[l]); }
  float s = 0.0f;
#pragma unroll
  for (int l = 0; l < LAGS; ++l) { v[l] = __expf(v[l] - m); s += v[l]; }
  const float inv = fast_rcp(s);
  float* wp = wlag + (size_t)p * LAGS;
#pragma unroll
  for (int l = 0; l < LAGS; ++l) wp[l] = v[l] * inv;
}

// ---------------------------------------------------------------------------
// Kernel 3: main fused kernel.
//   Block = 16(b) x 16(o) tile x one i-chunk of 32; 8 waves, 4 i per wave.
//   Per i: x_lagged tile via 3x V_WMMA_F32_16X16X4_F32 (K=11 padded to 12),
//   then spline-lerp + v_tanh_f32 MLP gate evaluated in the WMMA C/D layout
//   (lane = column o, VGPR r = batch row), accumulated over i in registers.
//   Deterministic LDS reduction across the 8 waves; partial per chunk.
// ---------------------------------------------------------------------------
__global__ __launch_bounds__(256) void cdkan_main(
    const float* __restrict__ x,       // (B,S,I)
    const float* __restrict__ coeffs,  // (O,I,8)
    const float* __restrict__ w1,      // (O,I,16)
    const float* __restrict__ b1,      // (O,I,16)
    const float* __restrict__ w2,      // (O,I,16)
    const float* __restrict__ b2,      // (O,I)
    const float* __restrict__ edge,    // (O,I)
    const float* __restrict__ xm,      // (B,I)  [ws]
    const float* __restrict__ wlag,    // (O,I,11) [ws]
    float* __restrict__ partial)       // (ICHUNKS,B,O) [ws]
{
  __shared__ float red[8][256];
  __shared__ float xm_s[16][IPC];      // 16 b-rows x 32 i of this chunk

  const int tid  = threadIdx.x;
  const int wave = tid >> 5;
  const int lane = tid & 31;
  const int n    = lane & 15;   // column index within tile (o) / A row (b)
  const int h2   = lane >> 4;   // half-wave selector

  const int chunk  = blockIdx.x >> 7;       // 0..3
  const int tileid = blockIdx.x & (TILES - 1);
  const int b0 = (tileid >> 3) * 16;        // 16 b-tiles
  const int o0 = (tileid & 7) * 16;         // 8 o-tiles
  const int o  = o0 + n;
  const int ibase = chunk * IPC;

  // stage xm tile: 512 floats, 2 per thread
#pragma unroll
  for (int e = tid * 2; e < tid * 2 + 2; ++e) {
    const int bl = e >> 5;         // 0..15
    const int il = e & 31;         // 0..31
    xm_s[bl][il] = xm[(size_t)(b0 + bl) * I_ + ibase + il];
  }
  __syncthreads();

  float acc[8];
#pragma unroll
  for (int r = 0; r < 8; ++r) acc[r] = 0.0f;

  for (int j = 0; j < IPC / 8; ++j) {        // 4 iterations
    const int il = wave * (IPC / 8) + j;     // i within chunk, 0..31
    const int i  = ibase + il;

    // ---- x_lagged 16x16 tile: A[m,k]=hist[b0+m, l, i], B[k,n]=w_lag[o0+n,i,l]
    // A layout (32-bit 16x4): m = lane&15, k = (lane>=16 ? 2 : 0) + v
    // hist[b,l,i] = x_history[b, S-1-l, i]
    const float* xrow = x + (size_t)(b0 + n) * S_ * I_ + i;
    const float* wrow = wlag + ((size_t)o * I_ + i) * LAGS;

    v8f c = {0.f, 0.f, 0.f, 0.f, 0.f, 0.f, 0.f, 0.f};
#pragma unroll
    for (int kc = 0; kc < 3; ++kc) {
      v2f a, bfrag;
#pragma unroll
      for (int v = 0; v < 2; ++v) {
        const int l = kc * 4 + h2 * 2 + v;   // global K index (lag)
        float av = 0.0f, bv = 0.0f;
        if (l < LAGS) {
          av = xrow[(size_t)(S_ - 1 - l) * I_];
          bv = wrow[l];
        }
        a[v] = av;
        bfrag[v] = bv;
      }
      c = __builtin_amdgcn_wmma_f32_16x16x4_f32(
          /*neg_a=*/false, a, /*neg_b=*/false, bfrag,
          /*c_mod=*/(short)0, c, /*reuse_a=*/false, /*reuse_b=*/false);
    }

    // ---- per-(o,i) parameters: constant across this lane's 8 batch rows
    const size_t oi = (size_t)o * I_ + i;
    float cf[CN];
#pragma unroll
    for (int k = 0; k < CN; ++k) cf[k] = coeffs[oi * CN + k];
    float mw1[H_], mb1v[H_], mw2[H_];
#pragma unroll
    for (int h = 0; h < H_; ++h) {
      mw1[h]  = w1[oi * H_ + h];
      mb1v[h] = b1[oi * H_ + h];
      mw2[h]  = w2[oi * H_ + h];
    }
    const float mb2  = b2[oi];
    const float mask = (edge[oi] > 0.0f) ? 1.0f : 0.0f;  // sigmoid(e)>0.5

    // ---- per-element spline + gate, in WMMA C layout (VGPR r -> batch row)
#pragma unroll
    for (int r = 0; r < 8; ++r) {
      const float s   = fast_sigmoid(c[r]);
      const float idx = s * GRIDM1;           // in [0, 4]
      int k = (int)idx;                       // trunc == floor (idx >= 0)
      k = min(k, CN - 1);
      const int k1 = min(k + 1, CN - 1);
      const float wf = idx - (float)k;
      const float y  = fmaf(cf[k1] - cf[k], wf, cf[k]);  // c0*(1-w)+c1*w

      const float xmv = xm_s[h2 * 8 + r][il];
      float asum = 0.0f;
#pragma unroll
      for (int h = 0; h < H_; ++h) {
        asum = fmaf(fast_tanh(fmaf(xmv, mw1[h], mb1v[h])), mw2[h], asum);
      }
      const float alpha = fast_sigmoid(asum + mb2);
      acc[r] = fmaf(y, alpha * mask, acc[r]);
    }
  }

  // ---- deterministic cross-wave reduction
#pragma unroll
  for (int r = 0; r < 8; ++r) red[wave][lane * 8 + r] = acc[r];
  __syncthreads();

  float total = 0.0f;
#pragma unroll
  for (int w = 0; w < 8; ++w) total += red[w][tid];

  const int lane2 = tid >> 3;           // original lane that produced element
  const int r2    = tid & 7;            // original VGPR index
  const int ob    = o0 + (lane2 & 15);
  const int bb    = b0 + (lane2 >> 4) * 8 + r2;
  partial[(size_t)chunk * B_ * O_ + (size_t)bb * O_ + ob] = total;
}

// ---------------------------------------------------------------------------
// Kernel 4: combine the 4 i-chunk partials (deterministic order)
// ---------------------------------------------------------------------------
__global__ __launch_bounds__(256) void combine_kernel(
    const float* __restrict__ partial, float* __restrict__ out) {
  const int t = blockIdx.x * blockDim.x + threadIdx.x;  // 0 .. B*O-1
  float s = 0.0f;
#pragma unroll
  for (int c = 0; c < ICHUNKS; ++c) s += partial[(size_t)c * B_ * O_ + t];
  out[t] = s;
}

// ---------------------------------------------------------------------------
extern "C" void kernel_launch(void* const* d_in, const int* in_sizes, int n_in,
                              void* d_out, int out_size, void* d_ws, size_t ws_size,
                              hipStream_t stream) {
  const float* x_history   = (const float*)d_in[0];  // (256,512,128)
  const float* coeffs      = (const float*)d_in[1];  // (128,128,8)
  const float* lag_logits  = (const float*)d_in[2];  // (128,128,11)
  const float* mod_w1      = (const float*)d_in[3];  // (128,128,16)
  const float* mod_b1      = (const float*)d_in[4];  // (128,128,16)
  const float* mod_w2      = (const float*)d_in[5];  // (128,128,16)
  const float* mod_b2      = (const float*)d_in[6];  // (128,128)
  const float* edge_logits = (const float*)d_in[7];  // (128,128)
  float* out = (float*)d_out;                        // (256,128)

  float* xm      = (float*)d_ws;                     // 32768 floats
  float* wlag    = xm + B_ * I_;                     // 180224 floats
  float* partial = wlag + O_ * I_ * LAGS;            // 4*32768 floats

  xm_kernel<<<B_, I_, 0, stream>>>(x_history, xm);
  softmax_kernel<<<(O_ * I_) / 128, 128, 0, stream>>>(lag_logits, wlag);
  cdkan_main<<<TILES * ICHUNKS, 256, 0, stream>>>(
      x_history, coeffs, mod_w1, mod_b1, mod_w2, mod_b2, edge_logits,
      xm, wlag, partial);
  combine_kernel<<<(B_ * O_) / 256, 256, 0, stream>>>(partial, out);
}